// OFTRotationModule_12635793785534
// MI455X (gfx1250) — compile-verified
//
#include <hip/hip_runtime.h>
#include <hip/hip_bf16.h>

typedef __attribute__((ext_vector_type(2))) float v2f;
typedef __attribute__((ext_vector_type(8))) float v8f;

#define BSZ 64            // rotation block size
#define RBLK 64           // number of rotation blocks
#define NELEM 2016        // strict upper-tri elements per block
#define NFEAT 4096        // in_features = RBLK * BSZ
#define NROWS 8192

// ---------------------------------------------------------------------------
// Kernel 1: Cayley-Neumann. One workgroup per rotation block r.
// Builds Q (skew-symmetric) in LDS, computes R = I + 2(Q + Q^2 + Q^3 + Q^4)
// via LDS ping-pong matmuls, stores R TRANSPOSED (Rt[c][k] = R[k][c]) so the
// WMMA B-operand in kernel 2 is a contiguous float2 per lane.
// ---------------------------------------------------------------------------
__global__ void oft_cayley_kernel(const float* __restrict__ w,
                                  float* __restrict__ Rt) {
  const int r   = blockIdx.x;
  const int tid = threadIdx.x;            // 0..255
  __shared__ float Q [BSZ * BSZ];
  __shared__ float P0[BSZ * BSZ];
  __shared__ float P1[BSZ * BSZ];

  const float* wr  = w + (size_t)r * NELEM;
  const int col  = tid & 63;              // element column handled by this thread
  const int row0 = tid >> 6;              // base row (stride 4 over e)

  // Build Q and P0 = Q
  #pragma unroll
  for (int e = 0; e < 16; ++e) {
    const int idx = tid + (e << 8);
    const int i = idx >> 6, j = idx & 63;
    float v = 0.0f;
    if (i < j)      v =  wr[i * 63 - ((i * (i - 1)) >> 1) + (j - i - 1)];
    else if (i > j) v = -wr[j * 63 - ((j * (j - 1)) >> 1) + (i - j - 1)];
    Q[idx]  = v;
    P0[idx] = v;
  }

  // racc = I + 2Q  (element (row0+4e, col))
  float racc[16];
  #pragma unroll
  for (int e = 0; e < 16; ++e) {
    const int idx = tid + (e << 8);
    racc[e] = (((idx >> 6) == (idx & 63)) ? 1.0f : 0.0f) + 2.0f * Q[idx];
  }
  __syncthreads();

  float* bufs[2] = {P0, P1};
  for (int term = 0; term < 3; ++term) {   // adds 2Q^2, 2Q^3, 2Q^4
    const float* src = bufs[term & 1];
    float*       dst = bufs[(term + 1) & 1];
    float acc[16];
    #pragma unroll
    for (int e = 0; e < 16; ++e) acc[e] = 0.0f;
    for (int t = 0; t < BSZ; ++t) {
      const float qv = Q[(t << 6) + col];
      #pragma unroll
      for (int e = 0; e < 16; ++e)
        acc[e] = fmaf(src[((row0 + (e << 2)) << 6) + t], qv, acc[e]);
    }
    #pragma unroll
    for (int e = 0; e < 16; ++e) {
      const int idx = ((row0 + (e << 2)) << 6) + col;
      dst[idx] = acc[e];
      racc[e] += 2.0f * acc[e];
    }
    __syncthreads();
  }

  // Store transposed: Rt[r][c][k] = R[k][c]
  float* o = Rt + (size_t)r * (BSZ * BSZ);
  #pragma unroll
  for (int e = 0; e < 16; ++e) {
    const int row = row0 + (e << 2);       // k
    o[(col << 6) + row] = racc[e];
  }
}

// ---------------------------------------------------------------------------
// Kernel 2: block-diagonal GEMM with V_WMMA_F32_16X16X4_F32.
// grid = (NROWS/256, RBLK), block = 256 threads (8 waves of 32).
// Each wave: 32 rows x 64 cols output for block r; K looped in chunks of 4.
// A operand layout (f32 16x4): lane L -> M = L&15, K = k0 + 2*(L>>4) + vgpr.
// B operand layout (f32 4x16): lane L -> N = L&15, K = k0 + 2*(L>>4) + vgpr.
// Both are a contiguous float2 given x row-major and Rt = R^T in LDS.
// launch_bounds(256,1): allow full VGPR budget -> no inner-loop scratch spill.
// ---------------------------------------------------------------------------
__global__ void __launch_bounds__(256, 1)
oft_apply_kernel(const float* __restrict__ x,
                 const float* __restrict__ Rt,
                 float* __restrict__ out) {
  const int r    = blockIdx.y;
  const int wave = threadIdx.x >> 5;       // 0..7
  const int lane = threadIdx.x & 31;
  const int lh   = lane >> 4;              // lane half
  const int ll   = lane & 15;
  const int row_base = blockIdx.x * 256 + wave * 32;
  const int colb = r * BSZ;

  __shared__ float Bt[BSZ * BSZ];          // Rt block: Bt[c*64 + k]
  {
    const float* src = Rt + (size_t)r * (BSZ * BSZ);
    #pragma unroll
    for (int e = 0; e < 16; ++e) {
      const int idx = threadIdx.x + (e << 8);
      Bt[idx] = src[idx];
    }
  }
  __syncthreads();

  v8f acc[2][4];
  #pragma unroll
  for (int mt = 0; mt < 2; ++mt)
    #pragma unroll
    for (int nt = 0; nt < 4; ++nt)
      #pragma unroll
      for (int i = 0; i < 8; ++i) acc[mt][nt][i] = 0.0f;

  const float* xrow0 = x + (size_t)(row_base + ll) * NFEAT + colb;
  const float* xrow1 = xrow0 + (size_t)16 * NFEAT;

  #pragma unroll 8
  for (int k0 = 0; k0 < BSZ; k0 += 4) {
    const int ka = k0 + 2 * lh;
    // x is streamed exactly once -> non-temporal loads
    const v2f a0 = __builtin_nontemporal_load((const v2f*)(xrow0 + ka));
    const v2f a1 = __builtin_nontemporal_load((const v2f*)(xrow1 + ka));
    #pragma unroll
    for (int nt = 0; nt < 4; ++nt) {
      const v2f b = *(const v2f*)(&Bt[((nt * 16 + ll) << 6) + ka]);
      acc[0][nt] = __builtin_amdgcn_wmma_f32_16x16x4_f32(
          false, a0, false, b, (short)0, acc[0][nt], false, false);
      acc[1][nt] = __builtin_amdgcn_wmma_f32_16x16x4_f32(
          false, a1, false, b, (short)0, acc[1][nt], false, false);
    }
  }

  // Store D: VGPR i -> M = i + 8*lh (per 16x16 f32 C/D layout), N = ll.
  // out is written exactly once -> non-temporal stores.
  float* obase = out + (size_t)row_base * NFEAT + colb;
  #pragma unroll
  for (int mt = 0; mt < 2; ++mt) {
    #pragma unroll
    for (int nt = 0; nt < 4; ++nt) {
      #pragma unroll
      for (int i = 0; i < 8; ++i) {
        const int m = mt * 16 + 8 * lh + i;
        __builtin_nontemporal_store(acc[mt][nt][i],
                                    obase + (size_t)m * NFEAT + nt * 16 + ll);
      }
    }
  }
}

extern "C" void kernel_launch(void* const* d_in, const int* in_sizes, int n_in,
                              void* d_out, int out_size, void* d_ws, size_t ws_size,
                              hipStream_t stream) {
  const float* x  = (const float*)d_in[0];      // (8192, 4096) fp32
  const float* w  = (const float*)d_in[1];      // (64, 2016) fp32
  float* out = (float*)d_out;                   // (8192, 4096) fp32
  float* Rt  = (float*)d_ws;                    // 64*64*64 floats = 1 MB

  oft_cayley_kernel<<<RBLK, 256, 0, stream>>>(w, Rt);

  dim3 grid(NROWS / 256, RBLK);
  oft_apply_kernel<<<grid, 256, 0, stream>>>(x, Rt, out);
}